// CompGCN_1202590843053
// MI455X (gfx1250) — compile-verified
//
#include <hip/hip_runtime.h>

#define N_NODES 50000
#define N_EDGES 640000
#define DCH     128
#define PITCH   132          // 128 + 4 pad: float4-aligned, bank-conflict-free column reads
#define EPSBN   1e-5f

typedef float v2f __attribute__((ext_vector_type(2)));
typedef float v8f __attribute__((ext_vector_type(8)));

// ---------------------------------------------------------------------------
// Stage a 128x128 row-major weight matrix (W[f][d]) into padded LDS.
// ---------------------------------------------------------------------------
__device__ __forceinline__ void stage_weights(const float* __restrict__ w, float* lw) {
    for (int i = threadIdx.x; i < (DCH * DCH) / 4; i += blockDim.x) {
        const int i4 = i * 4;
        const int r = i4 >> 7;          // row  (output feature f)
        const int c = i4 & 127;         // col  (input dim d), multiple of 4
        float4 v = ((const float4*)w)[i];
        *(float4*)&lw[r * PITCH + c] = v;   // (r*132+c)*4 bytes is 16B aligned
    }
}

// ---------------------------------------------------------------------------
// One WMMA macro-tile: 16 rows x 128 cols, K = 128 in steps of 4 (fp32 WMMA).
// A fragment: lane&15 = row, lane>>4 selects K pair {k0,k0+1} vs {k0+2,k0+3}.
// B fragment from LDS: b[j] = W[n][k0 + 2*kh + j]  (W row-major [f][d]).
// C/D: vgpr g holds out[m0+g+8*kh][n0 + (lane&15)].
// ---------------------------------------------------------------------------

// Edge kernel: per 16-edge tile (one wave), 8 waves / block, block stages
// weights. Phase 1: msg = (x[src]*ea) @ w_in^T, atomically scattered to agg.
// Phase 2: ea = ea @ w_rel^T + b_rel, in place.
__global__ __launch_bounds__(256) void edge_kernel(
    const float* __restrict__ x, float* __restrict__ ea,
    const float* __restrict__ w_in, const float* __restrict__ w_rel,
    const float* __restrict__ b_rel,
    const int* __restrict__ src, const int* __restrict__ dst,
    float* __restrict__ agg)
{
    __shared__ float lw[DCH * PITCH];

    const int wave = threadIdx.x >> 5;
    const int lane = threadIdx.x & 31;
    const int r16  = lane & 15;
    const int kh   = lane >> 4;
    const long e0  = (long)(blockIdx.x * 8 + wave) * 16;   // exact: 5000*8*16 = E
    const long eR  = e0 + r16;

    const long  xbase = (long)src[eR] * DCH;
    const float* earow = ea + eR * DCH;

    int drows[8];
#pragma unroll
    for (int g = 0; g < 8; ++g) drows[g] = dst[e0 + g + 8 * kh];

    const v8f vz = {0.f, 0.f, 0.f, 0.f, 0.f, 0.f, 0.f, 0.f};

    // ---------------- phase 1: message GEMM -------------------------------
    stage_weights(w_in, lw);
    __syncthreads();

    v8f acc[8];
#pragma unroll
    for (int t = 0; t < 8; ++t) acc[t] = vz;

    for (int k0 = 0; k0 < DCH; k0 += 4) {
        const int kk = k0 + 2 * kh;
        v2f xv = *(const v2f*)(x + xbase + kk);
        v2f ev = *(const v2f*)(earow + kk);
        v2f a;
        a.x = xv.x * ev.x;
        a.y = xv.y * ev.y;
#pragma unroll
        for (int t = 0; t < 8; ++t) {
            v2f b = *(const v2f*)&lw[(t * 16 + r16) * PITCH + kk];
            acc[t] = __builtin_amdgcn_wmma_f32_16x16x4_f32(
                false, a, false, b, (short)0, acc[t], false, false);
        }
    }

    // scatter-add messages into agg (L2-resident, fp32 atomics)
#pragma unroll
    for (int t = 0; t < 8; ++t) {
#pragma unroll
        for (int g = 0; g < 8; ++g) {
            atomicAdd(&agg[(long)drows[g] * DCH + t * 16 + r16], acc[t][g]);
        }
    }

    // ---------------- phase 2: relation GEMM (in place) -------------------
    __syncthreads();
    stage_weights(w_rel, lw);
    __syncthreads();

#pragma unroll
    for (int t = 0; t < 8; ++t) acc[t] = vz;

    for (int k0 = 0; k0 < DCH; k0 += 4) {
        const int kk = k0 + 2 * kh;
        v2f a = *(const v2f*)(earow + kk);
#pragma unroll
        for (int t = 0; t < 8; ++t) {
            v2f b = *(const v2f*)&lw[(t * 16 + r16) * PITCH + kk];
            acc[t] = __builtin_amdgcn_wmma_f32_16x16x4_f32(
                false, a, false, b, (short)0, acc[t], false, false);
        }
    }

#pragma unroll
    for (int t = 0; t < 8; ++t) {
        const float br = b_rel[t * 16 + r16];
#pragma unroll
        for (int g = 0; g < 8; ++g) {
            ea[(e0 + g + 8 * kh) * DCH + t * 16 + r16] = acc[t][g] + br;
        }
    }
}

// Node kernel: x_out = agg/max(cnt,1) + x @ w_self^T + b_self + [cnt>0]*b_in
__global__ __launch_bounds__(256) void node_kernel(
    const float* __restrict__ x, const float* __restrict__ agg,
    const float* __restrict__ cnt, const float* __restrict__ w_self,
    const float* __restrict__ b_self, const float* __restrict__ b_in,
    float* __restrict__ out)
{
    __shared__ float lw[DCH * PITCH];
    stage_weights(w_self, lw);
    __syncthreads();

    const int wave = threadIdx.x >> 5;
    const int lane = threadIdx.x & 31;
    const int r16  = lane & 15;
    const int kh   = lane >> 4;
    const int tile = blockIdx.x * 8 + wave;
    if (tile >= N_NODES / 16) return;          // wave-uniform: EXEC stays full
    const long n0 = (long)tile * 16;

    const float* xrow = x + (n0 + r16) * DCH;

    const v8f vz = {0.f, 0.f, 0.f, 0.f, 0.f, 0.f, 0.f, 0.f};
    v8f acc[8];
#pragma unroll
    for (int t = 0; t < 8; ++t) acc[t] = vz;

    for (int k0 = 0; k0 < DCH; k0 += 4) {
        const int kk = k0 + 2 * kh;
        v2f a = *(const v2f*)(xrow + kk);
#pragma unroll
        for (int t = 0; t < 8; ++t) {
            v2f b = *(const v2f*)&lw[(t * 16 + r16) * PITCH + kk];
            acc[t] = __builtin_amdgcn_wmma_f32_16x16x4_f32(
                false, a, false, b, (short)0, acc[t], false, false);
        }
    }

    float cv[8], inv[8];
#pragma unroll
    for (int g = 0; g < 8; ++g) {
        cv[g]  = cnt[n0 + g + 8 * kh];
        inv[g] = 1.f / fmaxf(cv[g], 1.f);
    }

#pragma unroll
    for (int t = 0; t < 8; ++t) {
        const int   nf = t * 16 + r16;
        const float bs = b_self[nf];
        const float bi = b_in[nf];
#pragma unroll
        for (int g = 0; g < 8; ++g) {
            const long m = n0 + g + 8 * kh;
            float v = acc[t][g] + agg[m * DCH + nf] * inv[g] + bs
                    + (cv[g] > 0.f ? bi : 0.f);
            out[m * DCH + nf] = v;
        }
    }
}

// In-degree counts
__global__ void count_kernel(const int* __restrict__ dst, float* __restrict__ cnt) {
    const int e = blockIdx.x * 256 + threadIdx.x;
    if (e < N_EDGES) atomicAdd(&cnt[dst[e]], 1.0f);
}

// BN statistics: stats[0..127]=sum, stats[128..255]=sumsq (per feature)
__global__ void bn_stats_kernel(const float* __restrict__ h, float* __restrict__ stats) {
    const int f = threadIdx.x & 127;
    const int r0 = blockIdx.x * 2 + (threadIdx.x >> 7);
    float s = 0.f, s2 = 0.f;
    for (int r = r0; r < N_NODES; r += gridDim.x * 2) {
        const float v = h[(long)r * DCH + f];
        s += v; s2 += v * v;
    }
    atomicAdd(&stats[f], s);
    atomicAdd(&stats[128 + f], s2);
}

// BN apply + ReLU
__global__ void bn_apply_kernel(const float* __restrict__ h,
                                const float* __restrict__ stats,
                                const float* __restrict__ gamma,
                                const float* __restrict__ beta,
                                float* __restrict__ xout)
{
    const int i = blockIdx.x * 256 + threadIdx.x;   // N*D / 256 exact
    const int f = i & 127;
    const float mu  = stats[f] * (1.0f / N_NODES);
    const float var = stats[128 + f] * (1.0f / N_NODES) - mu * mu;
    const float v = (h[i] - mu) * rsqrtf(var + EPSBN) * gamma[f] + beta[f];
    xout[i] = fmaxf(v, 0.f);
}

// ---------------------------------------------------------------------------
extern "C" void kernel_launch(void* const* d_in, const int* in_sizes, int n_in,
                              void* d_out, int out_size, void* d_ws, size_t ws_size,
                              hipStream_t stream)
{
    (void)in_sizes; (void)n_in; (void)out_size; (void)ws_size;

    const float* x_in   = (const float*)d_in[0];
    const float* ea_in  = (const float*)d_in[1];
    const float* w_self = (const float*)d_in[2];
    const float* b_self = (const float*)d_in[3];
    const float* w_in   = (const float*)d_in[4];
    const float* b_in   = (const float*)d_in[5];
    const float* w_rel  = (const float*)d_in[6];
    const float* b_rel  = (const float*)d_in[7];
    const float* gamma  = (const float*)d_in[8];
    const float* beta   = (const float*)d_in[9];
    const int*   ei     = (const int*)d_in[10];
    const int*   src    = ei;
    const int*   dst    = ei + N_EDGES;

    float* out    = (float*)d_out;
    float* ea_buf = out + (long)N_NODES * DCH;      // edge_attr lives in d_out

    const long ND = (long)N_NODES * DCH;
    float* ws      = (float*)d_ws;
    float* ws_x    = ws;               // current node features
    float* ws_h    = ws + ND;          // pre-BN activations
    float* ws_agg  = ws + 2 * ND;      // scatter accumulator
    float* ws_cnt  = ws + 3 * ND;      // in-degree
    float* ws_stat = ws + 3 * ND + N_NODES;

    // seed edge_attr into its final (in-place) location
    hipMemcpyAsync(ea_buf, ea_in, (size_t)N_EDGES * DCH * sizeof(float),
                   hipMemcpyDeviceToDevice, stream);

    // in-degree counts (once)
    hipMemsetAsync(ws_cnt, 0, N_NODES * sizeof(float), stream);
    count_kernel<<<(N_EDGES + 255) / 256, 256, 0, stream>>>(dst, ws_cnt);

    const float* xp = x_in;
    for (int l = 0; l < 3; ++l) {
        hipMemsetAsync(ws_agg, 0, ND * sizeof(float), stream);

        edge_kernel<<<N_EDGES / 128, 256, 0, stream>>>(
            xp, ea_buf,
            w_in + (long)l * DCH * DCH, w_rel + (long)l * DCH * DCH,
            b_rel + l * DCH, src, dst, ws_agg);

        float* hout = (l == 2) ? out : ws_h;
        node_kernel<<<(N_NODES / 16 + 7) / 8, 256, 0, stream>>>(
            xp, ws_agg, ws_cnt,
            w_self + (long)l * DCH * DCH, b_self + l * DCH, b_in + l * DCH,
            hout);

        if (l < 2) {
            hipMemsetAsync(ws_stat, 0, 256 * sizeof(float), stream);
            bn_stats_kernel<<<512, 256, 0, stream>>>(ws_h, ws_stat);
            bn_apply_kernel<<<(int)(ND / 256), 256, 0, stream>>>(
                ws_h, ws_stat, gamma + l * DCH, beta + l * DCH, ws_x);
            xp = ws_x;
        }
    }
}